// DEDCGCNEE_28776280883770
// MI455X (gfx1250) — compile-verified
//
#include <hip/hip_runtime.h>
#include <math.h>

// ---------------------------------------------------------------------------
// Problem: x (2,128,512,512) f32.
//   s = sum_c x            (2,512,512)   <- 256 MB read, bandwidth bound
//   e = 4x sobel3x3(s); edge = sum |e|   <- tiny, done on the WMMA unit
//   p = maxpool2x2(edge)   (2,256,256)
//   out[n,c,:,:] = p[n]    (2,128,256,256) <- 64 MB write
// ---------------------------------------------------------------------------

typedef float f32x4 __attribute__((ext_vector_type(4)));
typedef float v2f   __attribute__((ext_vector_type(2)));
typedef float v8f   __attribute__((ext_vector_type(8)));

#define IMG_W   512
#define IMG_HW  262144      // 512*512
#define NCH     128
#define POOL_HW 65536       // 256*256

// 4 Sobel kernels, taps k = dy*3+dx (cross-correlation, as lax.conv does),
// zero-padded to K=12 so three K=4 WMMA chunks cover them.
__constant__ float SOB[4][12] = {
    {-1.f, 0.f, 1.f, -2.f, 0.f, 2.f, -1.f, 0.f, 1.f, 0.f, 0.f, 0.f},
    { 1.f, 2.f, 1.f,  0.f, 0.f, 0.f, -1.f,-2.f,-1.f, 0.f, 0.f, 0.f},
    { 2.f, 1.f, 0.f,  1.f, 0.f,-1.f,  0.f,-1.f,-2.f, 0.f, 0.f, 0.f},
    { 0.f,-1.f,-2.f,  1.f, 0.f,-1.f,  2.f, 1.f, 0.f, 0.f, 0.f, 0.f},
};

// ------------------------- kernel 1: channel sum ---------------------------
// 131072 threads, one float4 of pixels each, 128 strided NT float4 loads.
// This is the 256 MB streaming read -> dominates runtime (~11 us at 23.3 TB/s).
__global__ void chan_sum_kernel(const float* __restrict__ x,
                                float* __restrict__ s) {
    const int tid  = blockIdx.x * blockDim.x + threadIdx.x;   // [0, 131072)
    const int nimg = tid >> 16;                               // / 65536
    const int pix4 = tid & 0xFFFF;
    const f32x4* xv = (const f32x4*)x;
    const long base = (long)nimg * (NCH * (IMG_HW / 4)) + pix4;

    f32x4 acc = {0.f, 0.f, 0.f, 0.f};
#pragma unroll 8
    for (int c = 0; c < NCH; ++c) {
        acc += __builtin_nontemporal_load(&xv[base + (long)c * (IMG_HW / 4)]);
    }
    ((f32x4*)s)[tid] = acc;   // s is tiny (2 MB), keep it cached (RT)
}

// -------------------- kernel 2: WMMA conv + abs + pool ---------------------
// A-fragment (16x4 f32, chunk q covers taps 4q..4q+3):
//   lanes 0-15 : M = lane, elem0 = K=4q,   elem1 = K=4q+1
//   lanes 16-31: M = lane-16, elem0 = K=4q+2, elem1 = K=4q+3
__device__ __forceinline__ v2f make_a(int lane, int q) {
    const int m     = lane & 15;
    const int kbase = 4 * q + ((lane & 16) ? 2 : 0);
    v2f a;
#pragma unroll
    for (int i = 0; i < 2; ++i) {
        const int k = kbase + i;                 // <= 11
        const float w = SOB[m & 3][k];
        a[i] = (m < 4 && k < 9) ? w : 0.0f;
    }
    return a;
}

__device__ __forceinline__ float load_s(const float* __restrict__ s,
                                        int nimg, int row, int col) {
    const bool ok = ((unsigned)row < 512u) && ((unsigned)col < 512u);
    const int rc = min(max(row, 0), 511);
    const int cc = min(max(col, 0), 511);
    const float v = s[nimg * IMG_HW + rc * IMG_W + cc];   // clamped -> always safe
    return ok ? v : 0.0f;                                  // v_cndmask, no branch
}

// B-fragment (4x16 f32): B[k][n] = s[h+dy(k)-1][w0+n+dx(k)-1] (SAME zero pad)
__device__ __forceinline__ v2f make_b(const float* __restrict__ s,
                                      int nimg, int h, int w0, int lane, int q) {
    const int n     = lane & 15;
    const int kbase = 4 * q + ((lane & 16) ? 2 : 0);
    v2f b;
#pragma unroll
    for (int i = 0; i < 2; ++i) {
        const int k  = kbase + i;
        const int kk = (k < 9) ? k : 8;
        const int dy = kk / 3, dx = kk - 3 * (kk / 3);
        const float v = load_s(s, nimg, h + dy - 1, w0 + n + dx - 1);
        b[i] = (k < 9) ? v : 0.0f;
    }
    return b;
}

// One wave -> conv rows {2r, 2r+1} x 16 cols -> 8 pooled outputs.
// 16384 waves total; 6 x v_wmma_f32_16x16x4_f32 per wave.
__global__ void conv_pool_kernel(const float* __restrict__ s,
                                 float* __restrict__ p) {
    const int wid  = (blockIdx.x * blockDim.x + threadIdx.x) >> 5;
    const int lane = threadIdx.x & 31;
    const int g    = wid & 31;            // w-group: 16 conv cols
    const int r    = (wid >> 5) & 255;    // pooled row
    const int nimg = wid >> 13;
    const int w0   = g * 16;

    const v2f a0 = make_a(lane, 0);
    const v2f a1 = make_a(lane, 1);
    const v2f a2 = make_a(lane, 2);

    float e[2];
#pragma unroll
    for (int rr = 0; rr < 2; ++rr) {
        const int h = 2 * r + rr;
        const v2f b0 = make_b(s, nimg, h, w0, lane, 0);
        const v2f b1 = make_b(s, nimg, h, w0, lane, 1);
        const v2f b2 = make_b(s, nimg, h, w0, lane, 2);
        v8f acc = {0.f, 0.f, 0.f, 0.f, 0.f, 0.f, 0.f, 0.f};
        acc = __builtin_amdgcn_wmma_f32_16x16x4_f32(false, a0, false, b0,
                                                    (short)0, acc, false, false);
        acc = __builtin_amdgcn_wmma_f32_16x16x4_f32(false, a1, false, b1,
                                                    (short)0, acc, false, false);
        acc = __builtin_amdgcn_wmma_f32_16x16x4_f32(false, a2, false, b2,
                                                    (short)0, acc, false, false);
        // D rows m=0..3 (lanes 0-15) = the 4 sobel responses for pixel n=lane
        e[rr] = fabsf(acc[0]) + fabsf(acc[1]) + fabsf(acc[2]) + fabsf(acc[3]);
    }

    // 2x2 maxpool: vertical in-lane, horizontal across lane pairs
    const float m = fmaxf(e[0], e[1]);
    const float o = __shfl_xor(m, 1, 32);
    const float pooled = fmaxf(m, o);
    if (lane < 16 && (lane & 1) == 0) {
        p[nimg * POOL_HW + r * 256 + (w0 >> 1) + (lane >> 1)] = pooled;
    }
}

// ---------------------- kernel 3: channel broadcast ------------------------
// 64 MB of NT float4 stores; p (512 KB) stays L2-resident across 128 re-reads.
__global__ void broadcast_kernel(const float* __restrict__ p,
                                 float* __restrict__ out) {
    const int tid   = blockIdx.x * blockDim.x + threadIdx.x;  // [0, 4194304)
    const int nimg  = tid >> 21;          // / (128*65536/4)
    const int pix4  = tid & 16383;        // float4 index within (256,256)
    const f32x4 v = ((const f32x4*)p)[nimg * (POOL_HW / 4) + pix4];
    __builtin_nontemporal_store(v, &((f32x4*)out)[tid]);
}

// ---------------------------------------------------------------------------
extern "C" void kernel_launch(void* const* d_in, const int* in_sizes, int n_in,
                              void* d_out, int out_size, void* d_ws, size_t ws_size,
                              hipStream_t stream) {
    (void)in_sizes; (void)n_in; (void)out_size; (void)ws_size;
    const float* x = (const float*)d_in[0];
    float* out = (float*)d_out;
    float* s = (float*)d_ws;              // 2*512*512 floats = 2 MB
    float* p = s + 2 * IMG_HW;            // 2*256*256 floats = 512 KB

    chan_sum_kernel <<<  512, 256, 0, stream>>>(x, s);
    conv_pool_kernel<<< 2048, 256, 0, stream>>>(s, p);
    broadcast_kernel<<<16384, 256, 0, stream>>>(p, out);
}